// PrototypeLayer_13993003450657
// MI455X (gfx1250) — compile-verified
//
#include <hip/hip_runtime.h>

typedef __attribute__((ext_vector_type(2))) float v2f;
typedef __attribute__((ext_vector_type(8))) float v8f;

#define T_TOK   (32 * 4096)          // 131072 tokens
#define DIMS    256
#define NPROTO  512
#define TDLL    ((size_t)T_TOK * DIMS)
#define ROWPITCH 260                 // dword pitch: mult of 4 (b128 align), 260%64=4 (no bank conflicts)
#define NBLOCKS (T_TOK / 32)         // 4096 blocks: 1 wave/block, 2x16-token tiles per wave

// ---------------------------------------------------------------- init: wsq[n] = ||w_n||^2
__global__ __launch_bounds__(512) void vq_init(const float* __restrict__ w,
                                               float* __restrict__ wsq) {
    int tid = threadIdx.x;
    if (tid < NPROTO) {
        const float4* wr = (const float4*)(w + (size_t)tid * DIMS);
        float s = 0.f;
#pragma unroll 8
        for (int i = 0; i < DIMS / 4; ++i) {
            float4 v = wr[i];
            s += v.x * v.x + v.y * v.y + v.z * v.z + v.w * v.w;
        }
        wsq[tid] = s;
    }
}

// ---------------------------------------------------------------- main fused kernel
// 1 wave per block; wave owns tokens [t0, t0+32): two 16x256 A tiles.
// Each B fragment (proto tile) load feeds TWO wmma's -> halves L2 traffic on w.
__global__ __launch_bounds__(32) void vq_main(const float* __restrict__ x,
                                              const float* __restrict__ w,
                                              const float* __restrict__ wsq,
                                              float* __restrict__ outProto,
                                              float* __restrict__ outX,
                                              float* __restrict__ partial) {
    __shared__ float xs[2 * 16 * ROWPITCH];  // 33,280 B: two padded 16x256 x tiles
    __shared__ int   idxs[32];               // winning prototype per token

    const int lane = threadIdx.x & 31;
    const int t0   = blockIdx.x * 32;

    // ---- stage both x tiles (32 rows x 256 f32) into LDS, float4 coalesced
#pragma unroll 4
    for (int m = 0; m < 32; ++m) {
#pragma unroll
        for (int i = 0; i < 2; ++i) {
            int k = (lane + i * 32) * 4;
            float4 v = *(const float4*)(x + (size_t)(t0 + m) * DIMS + k);
            *(float4*)(xs + (m >> 4) * (16 * ROWPITCH) + (m & 15) * ROWPITCH + k) = v;
        }
    }
    __syncthreads();

    // ---- phase 1: distances via f32 WMMA, per-lane running min over 32 N-tiles
    const int halfsel = lane >> 4;   // selects K sub-pair (A and B fragments)
    const int lidx    = lane & 15;   // A: row within tile / B: column n within tile
    float minval0[8], minval1[8];
    int   minidx0[8], minidx1[8];
#pragma unroll
    for (int j = 0; j < 8; ++j) {
        minval0[j] = 3.4e38f; minidx0[j] = 0;
        minval1[j] = 3.4e38f; minidx1[j] = 0;
    }

    const float* arow0 = xs + lidx * ROWPITCH + 2 * halfsel;
    const float* arow1 = arow0 + 16 * ROWPITCH;

    for (int nt = 0; nt < NPROTO / 16; ++nt) {
        const int n = nt * 16 + lidx;
        const float wsq_l = wsq[n];
        const float* wrow = w + (size_t)n * DIMS + 2 * halfsel;  // B: K pair of col n

        v8f c0 = {0.f, 0.f, 0.f, 0.f, 0.f, 0.f, 0.f, 0.f};
        v8f c1 = {0.f, 0.f, 0.f, 0.f, 0.f, 0.f, 0.f, 0.f};
#pragma unroll 4
        for (int kc = 0; kc < DIMS / 4; ++kc) {
            v2f b  = *(const v2f*)(wrow + 4 * kc);    // global b64, shared by both tiles
            v2f a0 = *(const v2f*)(arow0 + 4 * kc);   // ds_load_b64, conflict-free
            v2f a1 = *(const v2f*)(arow1 + 4 * kc);
            c0 = __builtin_amdgcn_wmma_f32_16x16x4_f32(false, a0, false, b,
                                                       (short)0, c0, false, false);
            c1 = __builtin_amdgcn_wmma_f32_16x16x4_f32(false, a1, false, b,
                                                       (short)0, c1, false, false);
        }
        // dist_rel[m][n] = ||w_n||^2 - 2 * x_m . w_n   (||x||^2 irrelevant to argmin)
#pragma unroll
        for (int j = 0; j < 8; ++j) {
            float d0 = wsq_l - 2.f * c0[j];
            if (d0 < minval0[j]) { minval0[j] = d0; minidx0[j] = n; }
            float d1 = wsq_l - 2.f * c1[j];
            if (d1 < minval1[j]) { minval1[j] = d1; minidx1[j] = n; }
        }
    }

    // ---- butterfly min across the 16 lanes of each half-wave (both tiles)
#pragma unroll
    for (int j = 0; j < 8; ++j) {
        float v0 = minval0[j]; int i0 = minidx0[j];
        float v1 = minval1[j]; int i1 = minidx1[j];
#pragma unroll
        for (int off = 1; off < 16; off <<= 1) {
            float ov0 = __shfl_xor(v0, off, 32); int oi0 = __shfl_xor(i0, off, 32);
            if (ov0 < v0) { v0 = ov0; i0 = oi0; }
            float ov1 = __shfl_xor(v1, off, 32); int oi1 = __shfl_xor(i1, off, 32);
            if (ov1 < v1) { v1 = ov1; i1 = oi1; }
        }
        minidx0[j] = i0; minidx1[j] = i1;
    }
    // C/D layout: VGPR j -> row j (lanes 0-15) / row j+8 (lanes 16-31)
    if (lidx == 0) {
#pragma unroll
        for (int j = 0; j < 8; ++j) {
            idxs[halfsel * 8 + j]      = minidx0[j];   // tokens t0 + 0..15
            idxs[16 + halfsel * 8 + j] = minidx1[j];   // tokens t0 + 16..31
        }
    }
    __syncthreads();

    // ---- phase 2: gather winning prototype rows, emit outputs, accumulate loss
    float lsum = 0.f;
#pragma unroll 2
    for (int m = 0; m < 32; ++m) {
        const int id = idxs[m];
        const float* wrow = w + (size_t)id * DIMS;
        const float* xrow = xs + (m >> 4) * (16 * ROWPITCH) + (m & 15) * ROWPITCH;
        const size_t ob = (size_t)(t0 + m) * DIMS;
#pragma unroll
        for (int i = 0; i < 2; ++i) {
            int k = (lane + i * 32) * 4;
            float4 xv = *(const float4*)(xrow + k);
            float4 p  = *(const float4*)(wrow + k);
            *(float4*)(outProto + ob + k) = p;                 // proto_st (forward) == proto
            float4 dv;
            dv.x = xv.x - p.x; dv.y = xv.y - p.y;
            dv.z = xv.z - p.z; dv.w = xv.w - p.w;
            *(float4*)(outX + ob + k) = dv;                    // x_out = x - proto
            lsum += dv.x * dv.x + dv.y * dv.y + dv.z * dv.z + dv.w * dv.w;
        }
    }
    // deterministic wave reduction (no float atomics)
#pragma unroll
    for (int off = 1; off < 32; off <<= 1) lsum += __shfl_xor(lsum, off, 32);
    if (lane == 0) partial[blockIdx.x] = lsum;
}

// ---------------------------------------------------------------- final loss reduction
__global__ __launch_bounds__(256) void vq_fin(const float* __restrict__ partial,
                                              float* __restrict__ lossOut) {
    __shared__ float sbuf[8];
    float s = 0.f;
    for (int i = threadIdx.x; i < NBLOCKS; i += 256) s += partial[i];
#pragma unroll
    for (int off = 1; off < 32; off <<= 1) s += __shfl_xor(s, off, 32);
    if ((threadIdx.x & 31) == 0) sbuf[threadIdx.x >> 5] = s;
    __syncthreads();
    if (threadIdx.x == 0) {
        float t = 0.f;
        for (int i = 0; i < 8; ++i) t += sbuf[i];
        // loss = q + 0.25*e, forward q == e == mean((proto - x)^2)
        *lossOut = 1.25f * t * (1.0f / (float)TDLL);
    }
}

extern "C" void kernel_launch(void* const* d_in, const int* in_sizes, int n_in,
                              void* d_out, int out_size, void* d_ws, size_t ws_size,
                              hipStream_t stream) {
    const float* x = (const float*)d_in[0];       // (32,4096,256) f32
    const float* w = (const float*)d_in[1];       // (512,256) f32
    float* out      = (float*)d_out;              // [proto_st | x_out | loss]
    float* outProto = out;
    float* outX     = out + TDLL;
    float* lossOut  = out + 2 * TDLL;

    float* wsq     = (float*)d_ws;                // 512 floats
    float* partial = (float*)d_ws + NPROTO;       // 4096 floats

    vq_init<<<1, 512, 0, stream>>>(w, wsq);
    vq_main<<<NBLOCKS, 32, 0, stream>>>(x, w, wsq, outProto, outX, partial);
    vq_fin<<<1, 256, 0, stream>>>(partial, lossOut);
}